// LinearGMMInterpolation_37684043055764
// MI455X (gfx1250) — compile-verified
//
#include <hip/hip_runtime.h>
#include <math.h>

typedef __attribute__((ext_vector_type(2))) float v2f;
typedef __attribute__((ext_vector_type(8))) float v8f;

// workspace/LDS float offsets
#define GOFF   0      // 64 x 8 stacked G_k rows (row r = 8k+i)
#define LAMOFF 512    // 8 x 8 (lambda_i - 1)
#define ASOFF  576    // 8 x 8 aS = G*muS
#define DAOFF  640    // 8 x 8 (G*muE - G*muS)
#define LDOFF  704    // 8 logdet(S_k)
#define SWOFF  712    // 8 start weights (LDS only)
#define EWOFF  720    // 8 end weights   (LDS only)
#define SHTOT  728

// ---------------- Stage 0: per-component simultaneous diagonalization ----------------
__global__ void gmm_setup(const float* __restrict__ sm, const float* __restrict__ em,
                          const float* __restrict__ sc, const float* __restrict__ ec,
                          float* __restrict__ ws)
{
    int k = threadIdx.x;
    if (k >= 8) return;

    float S[8][8], E[8][8];
    for (int i = 0; i < 8; ++i)
        for (int j = 0; j < 8; ++j) {
            S[i][j] = sc[k * 64 + i * 8 + j];
            E[i][j] = ec[k * 64 + i * 8 + j];
        }

    // Cholesky: S = C C^T (C lower)
    float C[8][8];
    for (int i = 0; i < 8; ++i)
        for (int j = 0; j < 8; ++j) C[i][j] = 0.0f;
    for (int j = 0; j < 8; ++j) {
        float s = S[j][j];
        for (int q = 0; q < j; ++q) s -= C[j][q] * C[j][q];
        float d = sqrtf(fmaxf(s, 1e-20f));
        C[j][j] = d;
        float inv = 1.0f / d;
        for (int i = j + 1; i < 8; ++i) {
            float v = S[i][j];
            for (int q = 0; q < j; ++q) v -= C[i][q] * C[j][q];
            C[i][j] = v * inv;
        }
    }

    // Ci = C^{-1} (lower triangular inverse)
    float Ci[8][8];
    for (int i = 0; i < 8; ++i)
        for (int j = 0; j < 8; ++j) Ci[i][j] = 0.0f;
    for (int j = 0; j < 8; ++j) {
        Ci[j][j] = 1.0f / C[j][j];
        for (int i = j + 1; i < 8; ++i) {
            float s = 0.0f;
            for (int q = j; q < i; ++q) s += C[i][q] * Ci[q][j];
            Ci[i][j] = -s / C[i][i];
        }
    }

    // M = Ci * E * Ci^T  (symmetric PD)
    float T[8][8], Mm[8][8];
    for (int i = 0; i < 8; ++i)
        for (int j = 0; j < 8; ++j) {
            float s = 0.0f;
            for (int q = 0; q < 8; ++q) s += Ci[i][q] * E[q][j];
            T[i][j] = s;
        }
    for (int i = 0; i < 8; ++i)
        for (int j = 0; j < 8; ++j) {
            float s = 0.0f;
            for (int q = 0; q < 8; ++q) s += T[i][q] * Ci[j][q];
            Mm[i][j] = s;
        }

    // Cyclic Jacobi eigendecomposition: M = Q diag(lam) Q^T
    float Q[8][8];
    for (int i = 0; i < 8; ++i)
        for (int j = 0; j < 8; ++j) Q[i][j] = (i == j) ? 1.0f : 0.0f;
    for (int sweep = 0; sweep < 12; ++sweep) {
        for (int p = 0; p < 7; ++p) {
            for (int q = p + 1; q < 8; ++q) {
                float apq = Mm[p][q];
                if (fabsf(apq) <= 1e-13f) continue;
                float theta = (Mm[q][q] - Mm[p][p]) / (2.0f * apq);
                float tt = copysignf(1.0f, theta) / (fabsf(theta) + sqrtf(1.0f + theta * theta));
                float cc = 1.0f / sqrtf(1.0f + tt * tt);
                float ss = tt * cc;
                for (int r = 0; r < 8; ++r) {
                    float u = Mm[r][p], v = Mm[r][q];
                    Mm[r][p] = cc * u - ss * v; Mm[r][q] = ss * u + cc * v;
                }
                for (int r = 0; r < 8; ++r) {
                    float u = Mm[p][r], v = Mm[q][r];
                    Mm[p][r] = cc * u - ss * v; Mm[q][r] = ss * u + cc * v;
                }
                for (int r = 0; r < 8; ++r) {
                    float u = Q[r][p], v = Q[r][q];
                    Q[r][p] = cc * u - ss * v; Q[r][q] = ss * u + cc * v;
                }
            }
        }
    }

    // G = Q^T * Ci
    float G[8][8];
    for (int i = 0; i < 8; ++i)
        for (int j = 0; j < 8; ++j) {
            float s = 0.0f;
            for (int q = 0; q < 8; ++q) s += Q[q][i] * Ci[q][j];
            G[i][j] = s;
        }

    float ld = 0.0f;
    for (int i = 0; i < 8; ++i) ld += logf(C[i][i]);

    for (int i = 0; i < 8; ++i) {
        float as = 0.0f, ae = 0.0f;
        for (int j = 0; j < 8; ++j) {
            as += G[i][j] * sm[k * 8 + j];
            ae += G[i][j] * em[k * 8 + j];
            ws[GOFF + (k * 8 + i) * 8 + j] = G[i][j];
        }
        ws[LAMOFF + k * 8 + i] = Mm[i][i] - 1.0f;   // lambda_i - 1
        ws[ASOFF + k * 8 + i] = as;
        ws[DAOFF + k * 8 + i] = ae - as;
    }
    ws[LDOFF + k] = 2.0f * ld;
}

// ---------------- Stage 1: WMMA GEMM Z^T = G_stacked * X^T + per-point epilogue ----------------
__global__ __launch_bounds__(256) void gmm_main(const float* __restrict__ x,
                                                const float* __restrict__ t,
                                                const float* __restrict__ sw8,
                                                const float* __restrict__ ew8,
                                                const float* __restrict__ ws,
                                                float* __restrict__ out, int Bn)
{
    __shared__ __align__(16) float sh[SHTOT];
    int tid = threadIdx.x;
    for (int i = tid; i < LDOFF; i += 256) sh[i] = ws[i];
    if (tid < 8) {
        sh[LDOFF + tid] = ws[LDOFF + tid];
        sh[SWOFF + tid] = sw8[tid];
        sh[EWOFF + tid] = ew8[tid];
    }
    __syncthreads();

    int lane = tid & 31;
    int wave = tid >> 5;
    int half = lane >> 4;      // 0: even components / K={0,1}; 1: odd / K={2,3}
    int p    = lane & 15;      // point within tile == matrix row within tile
    int co   = 2 * half;       // K-pair base within 4-wide K step

    long base = ((long)blockIdx.x * 8 + wave) * 16;
    long pid  = base + p;
    long pidc = (pid < (long)Bn) ? pid : (long)(Bn - 1);

    // A operand (G_stacked, 64x8): tile tau rows 16tau..16tau+15, K-step s cols 4s..4s+3
    v2f Ag[4][2];
#pragma unroll
    for (int tau = 0; tau < 4; ++tau)
#pragma unroll
        for (int s = 0; s < 2; ++s)
            Ag[tau][s] = *(const v2f*)&sh[GOFF + (16 * tau + p) * 8 + 4 * s + co];

    // B operand (X^T, 8xN): lane p holds column p, K pair {4s+co, 4s+co+1}
    const float* xr = x + pidc * 8;
    v2f Bx[2];
#pragma unroll
    for (int s = 0; s < 2; ++s)
        Bx[s] = *(const v2f*)&xr[4 * s + co];

    // 4 M-tiles x 2 K-steps of V_WMMA_F32_16X16X4_F32
    v8f acc[4];
#pragma unroll
    for (int tau = 0; tau < 4; ++tau) {
        v8f c = {0.f, 0.f, 0.f, 0.f, 0.f, 0.f, 0.f, 0.f};
        c = __builtin_amdgcn_wmma_f32_16x16x4_f32(false, Ag[tau][0], false, Bx[0],
                                                  (short)0, c, false, false);
        c = __builtin_amdgcn_wmma_f32_16x16x4_f32(false, Ag[tau][1], false, Bx[1],
                                                  (short)0, c, false, false);
        acc[tau] = c;
    }

    // Epilogue: lane holds all 8 z-rows of component k = 2*tau + half, for point p.
    float tv = t[pidc];
    float av[4];
#pragma unroll
    for (int tau = 0; tau < 4; ++tau) {
        int k = 2 * tau + half;
        float maha = 0.0f, prod = 1.0f;
#pragma unroll
        for (int i = 0; i < 8; ++i) {
            float z = acc[tau][i];
            float u = z - (sh[ASOFF + k * 8 + i] + tv * sh[DAOFF + k * 8 + i]);
            float d = 1.0f + tv * sh[LAMOFF + k * 8 + i];   // (1-t) + t*lambda_i
            maha += u * u / d;
            prod *= d;
        }
        float logdet = sh[LDOFF + k] + __logf(prod);
        float lp = -0.5f * (maha + logdet + 8.0f * 1.83787706640934534f); // D*log(2pi)
        float wk = sh[SWOFF + k] + tv * (sh[EWOFF + k] - sh[SWOFF + k]);
        av[tau] = __logf(wk) + lp;
    }

    // logsumexp over 4 local components, then merge halves via ds_swizzle (xor 16)
    float m = fmaxf(fmaxf(av[0], av[1]), fmaxf(av[2], av[3]));
    float s4 = __expf(av[0] - m) + __expf(av[1] - m) + __expf(av[2] - m) + __expf(av[3] - m);
    float m2 = __int_as_float(__builtin_amdgcn_ds_swizzle(__float_as_int(m),  0x401F));
    float s2 = __int_as_float(__builtin_amdgcn_ds_swizzle(__float_as_int(s4), 0x401F));
    float M  = fmaxf(m, m2);
    float tot = s4 * __expf(m - M) + s2 * __expf(m2 - M);

    float sumw = 0.0f;
#pragma unroll
    for (int k = 0; k < 8; ++k)
        sumw += sh[SWOFF + k] + tv * (sh[EWOFF + k] - sh[SWOFF + k]);

    float res = M + __logf(tot) - __logf(sumw);
    if (half == 0 && pid < (long)Bn) out[pid] = res;
}

extern "C" void kernel_launch(void* const* d_in, const int* in_sizes, int n_in,
                              void* d_out, int out_size, void* d_ws, size_t ws_size,
                              hipStream_t stream) {
    const float* x  = (const float*)d_in[0];
    const float* t  = (const float*)d_in[1];
    const float* sm = (const float*)d_in[2];
    const float* em = (const float*)d_in[3];
    const float* sc = (const float*)d_in[4];
    const float* ec = (const float*)d_in[5];
    const float* sw = (const float*)d_in[6];
    const float* ew = (const float*)d_in[7];
    float* out = (float*)d_out;
    float* ws  = (float*)d_ws;

    int Bn = in_sizes[0] / 8;

    gmm_setup<<<1, 32, 0, stream>>>(sm, em, sc, ec, ws);

    int tiles  = (Bn + 15) / 16;
    int blocks = (tiles + 7) / 8;   // 8 waves per block, 16 points per wave
    gmm_main<<<blocks, 256, 0, stream>>>(x, t, sw, ew, ws, out, Bn);
}